// ScaledDotProductAttention_24300924961205
// MI455X (gfx1250) — compile-verified
//
#include <hip/hip_runtime.h>
#include <stdint.h>

// ---------------------------------------------------------------------------
// Scaled dot-product attention (B=64, S=2048, D=64, fp32) for gfx1250.
//  - V_WMMA_F32_16X16X4_F32 for q@k^T and attn@v
//  - TDM tensor_load_to_lds (TENSORcnt) for K/V tile staging, with HW padding
//    producing a bank-conflict-free 68-float LDS row stride
//  - non-temporal streaming stores for the 1 GB attn tensor
// ---------------------------------------------------------------------------

typedef float v2f   __attribute__((ext_vector_type(2)));
typedef float v8f   __attribute__((ext_vector_type(8)));
typedef unsigned int u32x4 __attribute__((ext_vector_type(4)));
typedef int   i32x4 __attribute__((ext_vector_type(4)));
typedef int   i32x8 __attribute__((ext_vector_type(8)));

#define B_  64
#define S_  2048
#define D_  64
#define NTILE (S_ / 16)
#define SCALE 0.125f            // 1 / temperature (= 1/sqrt(64))
#define NEG_INF (-__builtin_inff())
#define KSTRIDE 68              // LDS row stride (floats): 64 data + 4 TDM pad
#define PSTRIDE 17              // padded LDS row stride for P-transpose scratch

__device__ __forceinline__ v8f wmma_f32_k4(v2f a, v2f b, v8f c) {
  // D = A(16x4 f32) * B(4x16 f32) + C(16x16 f32)
  return __builtin_amdgcn_wmma_f32_16x16x4_f32(false, a, false, b, (short)0, c,
                                               false, false);
}

// TDM: DMA one 16x64 fp32 tile (row stride 64 floats in memory) into LDS,
// inserting 4 DWORDs of padding after every 64 DWORDs -> LDS stride 68 floats.
__device__ __forceinline__ void tdm_load_tile(const float* gsrc, float* lds_dst) {
  const uint64_t ga = (uint64_t)(uintptr_t)gsrc;
  const uint32_t la = (uint32_t)(uintptr_t)lds_dst; // LDS address: low 32 bits of the pointer

  u32x4 g0;
  g0[0] = 1u;                                        // count=1 (valid user D#)
  g0[1] = la;                                        // lds_addr [63:32]
  g0[2] = (uint32_t)ga;                              // global_addr[31:0]
  g0[3] = (uint32_t)((ga >> 32) & 0x1FFFFFFu)        // global_addr[56:32]
        | (2u << 30);                                // type = 2 ("image")

  i32x8 g1;
  g1[0] = (2 << 16)                                  // data_size = 4 bytes
        | (1 << 20)                                  // pad_enable
        | (5 << 22)                                  // pad_interval: 64 DWORDs
        | (3 << 25);                                 // pad_amount: 4 DWORDs
  g1[1] = (int)(64u << 16);                          // tensor_dim0 = 64 (lo16)
  g1[2] = (int)(2048u << 16);                        // dim0 hi | tensor_dim1 lo = 2048
  g1[3] = (int)(64u << 16);                          // dim1 hi | tile_dim0 = 64
  g1[4] = 16;                                        // tile_dim1 = 16, tile_dim2 = 0
  g1[5] = 64;                                        // tensor_dim0_stride = 64
  g1[6] = 0;
  g1[7] = 0;

  i32x4 z4 = {0, 0, 0, 0};                           // groups 2/3 unused (2-D)
  i32x8 z8 = {0, 0, 0, 0, 0, 0, 0, 0};               // extra group (clang-23 form)
  __builtin_amdgcn_tensor_load_to_lds(g0, g1, z4, z4, z8, 0);
}

__global__ __launch_bounds__(256, 2)
void sdpa_wmma_f32_kernel(const float* __restrict__ q,
                          const float* __restrict__ k,
                          const float* __restrict__ v,
                          const uint8_t* __restrict__ mask,
                          float* __restrict__ out,
                          float* __restrict__ attn)
{
  __shared__ float lk[2][16 * KSTRIDE];   // double-buffered K tile (16 x 64)
  __shared__ float lv[2][16 * KSTRIDE];   // double-buffered V tile (16 x 64)
  __shared__ float lp[8][16 * PSTRIDE];   // per-wave P-tile transpose scratch

  const int tid   = threadIdx.x;
  const int lane  = tid & 31;
  const int wave  = tid >> 5;
  const int lhalf = lane >> 4;      // 0: lanes 0-15, 1: lanes 16-31
  const int l16   = lane & 15;
  const bool leader = (wave == 0);  // wave that drives the TDM

  const int b  = blockIdx.x >> 4;   // 16 q-chunks (of 128 rows) per batch
  const int qc = blockIdx.x & 15;
  const int q0 = (qc * 8 + wave) * 16;   // this wave's 16 query rows

  const size_t qbase = ((size_t)b * S_ + q0) * D_;
  const size_t kbase = (size_t)b * S_ * D_;
  const size_t mbase = (size_t)b * S_ * S_;   // mask / attn row-block base

  // ---- Q tile in WMMA A-layout: lane m<16 holds (M=m, K=4s..4s+1),
  //      lanes 16-31 hold (M=m-16, K=4s+2..4s+3), s = 0..15 ----
  v2f qa[16];
  {
    const float* qrow = q + qbase + (size_t)l16 * D_ + 2 * lhalf;
    #pragma unroll
    for (int s = 0; s < 16; ++s) {
      qa[s].x = qrow[4 * s + 0];
      qa[s].y = qrow[4 * s + 1];
    }
  }

  // ---- scores tile: S[16q x 16k] = Q(16x64) x K_tile^T(64x16), K dim = d ----
  auto compute_scores = [&](int buf) -> v8f {
    v8f c = {};
    const float* kb = &lk[buf][(size_t)l16 * KSTRIDE + 2 * lhalf];
    #pragma unroll
    for (int s = 0; s < 16; ++s) {
      v2f bb;                       // B-layout: lane = N (k-row), regs/halves = K (d)
      bb.x = kb[4 * s + 0];
      bb.y = kb[4 * s + 1];
      c = wmma_f32_k4(qa[s], bb, c);
    }
    return c;
  };

  // =======================================================================
  // Pass 1: online softmax stats (lane-local, merged across lanes at end)
  // =======================================================================
  float m[8], l[8];
  #pragma unroll
  for (int r = 0; r < 8; ++r) { m[r] = NEG_INF; l[r] = 0.0f; }

  if (leader) {
    tdm_load_tile(k + kbase, &lk[0][0]);
    __builtin_amdgcn_s_wait_tensorcnt(0);
  }
  __syncthreads();

  for (int t = 0; t < NTILE; ++t) {
    const int buf = t & 1;
    if (leader && (t + 1 < NTILE))
      tdm_load_tile(k + kbase + (size_t)(t + 1) * 16 * D_, &lk[buf ^ 1][0]);

    v8f c = compute_scores(buf);

    const int kc = t * 16 + l16;
    const uint8_t* mp = mask + mbase + (size_t)(q0 + 8 * lhalf) * S_ + kc;
    #pragma unroll
    for (int r = 0; r < 8; ++r) {           // C/D layout: reg r -> row r (+8 high half)
      float sc = c[r] * SCALE;
      if (mp[(size_t)r * S_]) sc = NEG_INF;
      const float mo = m[r];
      const float mn = fmaxf(mo, sc);
      const float a  = (mo > NEG_INF) ? __expf(mo - mn) : 0.0f;
      const float e  = (sc > NEG_INF) ? __expf(sc - mn) : 0.0f;
      l[r] = l[r] * a + e;
      m[r] = mn;
    }
    if (leader) __builtin_amdgcn_s_wait_tensorcnt(0);   // next tile landed
    __syncthreads();
  }

  // merge (m,l) across the 16 lanes sharing each row (xor stays inside halves)
  #pragma unroll
  for (int r = 0; r < 8; ++r) {
    float mm = m[r], ll = l[r];
    #pragma unroll
    for (int off = 1; off < 16; off <<= 1) {
      const float mo = __shfl_xor(mm, off, 32);
      const float lo = __shfl_xor(ll, off, 32);
      const float mn = fmaxf(mm, mo);
      const float a  = (mm > NEG_INF) ? __expf(mm - mn) : 0.0f;
      const float bb = (mo > NEG_INF) ? __expf(mo - mn) : 0.0f;
      ll = ll * a + lo * bb;
      mm = mn;
    }
    m[r] = mm;
    l[r] = 1.0f / ll;     // keep reciprocal of row sum
  }

  // =======================================================================
  // Pass 2: recompute scores, write normalized attn (NT stores), O += P x V
  // =======================================================================
  v8f acc[4] = {};                 // O tile: 16q x 64d as 4 N-blocks of 16
  float* myp = &lp[wave][0];

  if (leader) {
    tdm_load_tile(k + kbase, &lk[0][0]);
    tdm_load_tile(v + kbase, &lv[0][0]);
    __builtin_amdgcn_s_wait_tensorcnt(0);
  }
  __syncthreads();

  for (int t = 0; t < NTILE; ++t) {
    const int buf = t & 1;
    if (leader && (t + 1 < NTILE)) {
      tdm_load_tile(k + kbase + (size_t)(t + 1) * 16 * D_, &lk[buf ^ 1][0]);
      tdm_load_tile(v + kbase + (size_t)(t + 1) * 16 * D_, &lv[buf ^ 1][0]);
    }

    v8f c = compute_scores(buf);

    const int kc = t * 16 + l16;
    const uint8_t* mp = mask + mbase + (size_t)(q0 + 8 * lhalf) * S_ + kc;
    float* ap = attn + mbase + (size_t)(q0 + 8 * lhalf) * S_ + kc;
    #pragma unroll
    for (int r = 0; r < 8; ++r) {
      float sc = c[r] * SCALE;
      if (mp[(size_t)r * S_]) sc = NEG_INF;
      const float e = (sc > NEG_INF) ? __expf(sc - m[r]) : 0.0f;
      const float p = e * l[r];
      __builtin_nontemporal_store(p, ap + (size_t)r * S_);   // stream 1GB attn
      myp[(r + 8 * lhalf) * PSTRIDE + l16] = p;              // P^T staging
    }
    asm volatile("" ::: "memory");   // keep LDS stores before wave-local reloads

    // O(16x64) += P(16x16) x V_tile(16x64); K dim = 16 k-rows -> 4 wmma steps
    #pragma unroll
    for (int s = 0; s < 4; ++s) {
      v2f a;                         // A-layout: lane = M (q row), halves split K
      a.x = myp[l16 * PSTRIDE + 4 * s + 2 * lhalf + 0];
      a.y = myp[l16 * PSTRIDE + 4 * s + 2 * lhalf + 1];
      const float* vb0 = &lv[buf][(size_t)(4 * s + 2 * lhalf + 0) * KSTRIDE + l16];
      const float* vb1 = &lv[buf][(size_t)(4 * s + 2 * lhalf + 1) * KSTRIDE + l16];
      #pragma unroll
      for (int nb = 0; nb < 4; ++nb) {
        v2f bb;                      // B-layout: lane = N (d col), K rows via regs/halves
        bb.x = vb0[nb * 16];
        bb.y = vb1[nb * 16];
        acc[nb] = wmma_f32_k4(a, bb, acc[nb]);
      }
    }
    if (leader) __builtin_amdgcn_s_wait_tensorcnt(0);   // next K/V tiles landed
    __syncthreads();
  }

  // ---- store O tile: lane = d col within block, reg = q row ----
  #pragma unroll
  for (int nb = 0; nb < 4; ++nb) {
    #pragma unroll
    for (int r = 0; r < 8; ++r) {
      const size_t o = ((size_t)b * S_ + q0 + r + 8 * lhalf) * D_ + nb * 16 + l16;
      out[o] = acc[nb][r];
    }
  }
}

extern "C" void kernel_launch(void* const* d_in, const int* in_sizes, int n_in,
                              void* d_out, int out_size, void* d_ws, size_t ws_size,
                              hipStream_t stream) {
  (void)in_sizes; (void)n_in; (void)d_ws; (void)ws_size; (void)out_size;
  const float*   q    = (const float*)d_in[0];
  const float*   k    = (const float*)d_in[1];
  const float*   v    = (const float*)d_in[2];
  const uint8_t* mask = (const uint8_t*)d_in[3];   // jnp bool -> 1 byte

  float* out  = (float*)d_out;                          // B*S*D floats
  float* attn = out + (size_t)B_ * S_ * D_;             // B*S*S floats

  // grid: 64 batches x 16 q-chunks; block: 8 waves, each owns 16 q rows
  dim3 grid(B_ * (S_ / (16 * 8)));
  dim3 block(256);
  sdpa_wmma_f32_kernel<<<grid, block, 0, stream>>>(q, k, v, mask, out, attn);
}